// SignedGraph_6949257085152
// MI455X (gfx1250) — compile-verified
//
#include <hip/hip_runtime.h>
#include <hip/hip_bf16.h>
#include <math.h>

// ---------------------------------------------------------------------------
// SignedGraph attention on MI455X (gfx1250, wave32).
//
// Algebraic reductions (exact in fp32):
//   E = node@W^T + b is only consumed via dot products with w_att halves
//     -> s_self  = node @ u_self  + c_self,   u_self  = W^T w_att[de:]
//     -> s_other = node @ u_other + c_other+b_att
//   edge[i,j] = s_self[i] + s_other[j]  (rank-1)
//   softmax is per-column (axis=0) -> per-column (max, 1/sum(exp)) stats
//   relu(alpha * node[j]) = alpha * relu(node[j])  (alpha > 0)
//     -> out = (pos_alpha + neg_alpha) @ relu(node)   [1024x1024x128 f32 GEMM]
//
// GEMM: fp32 WMMA (V_WMMA_F32_16X16X4_F32) with the B panel (node) staged into
// LDS through the gfx1250 async-to-LDS path (global_load_async_to_lds_b128,
// ASYNCcnt), double-buffered in 64-row chunks and overlapped with the
// exp-heavy alpha-tile phase.
// ---------------------------------------------------------------------------

typedef __attribute__((ext_vector_type(2))) float v2f;
typedef __attribute__((ext_vector_type(8))) float v8f;

#define NNODES 1024
#define DN 128
#define NEGC (-1e8f)

__device__ __forceinline__ float leaky(float x) { return x >= 0.f ? x : 0.2f * x; }

// ws float layout:
//  [0,128)   u_self        [128,256) u_other     [256] c_self  [257] c_other+b_att
//  [512,1536)  s_self      [1536,2560) s_other
//  [2560,3584) M_pos       [3584,4608) invD_pos
//  [4608,5632) M_neg       [5632,6656) invD_neg

// --------------------------- kernel 1: fold weights -------------------------
__global__ __launch_bounds__(128) void prep_kernel(
    const float* __restrict__ W_emb, const float* __restrict__ b_emb,
    const float* __restrict__ w_att, const float* __restrict__ b_att,
    float* __restrict__ ws) {
  int k = threadIdx.x;  // 0..127
  float us = 0.f, uo = 0.f;
  for (int d = 0; d < DN; ++d) {
    float w = W_emb[d * DN + k];
    uo += w * w_att[d];
    us += w * w_att[DN + d];
  }
  ws[k] = us;
  ws[DN + k] = uo;
  if (k == 0) {
    float cs = 0.f, co = 0.f;
    for (int d = 0; d < DN; ++d) {
      cs += b_emb[d] * w_att[DN + d];
      co += b_emb[d] * w_att[d];
    }
    ws[256] = cs;
    ws[257] = co + b_att[0];
  }
}

// --------------------------- kernel 2: row scores (GEMV) --------------------
__global__ __launch_bounds__(256) void score_kernel(
    const float* __restrict__ node, const float* __restrict__ ws,
    float* __restrict__ s_self, float* __restrict__ s_other) {
  int row = blockIdx.x * 8 + (threadIdx.x >> 5);
  int lane = threadIdx.x & 31;
  float as = 0.f, ao = 0.f;
  for (int k = lane; k < DN; k += 32) {
    float x = node[row * DN + k];
    as += x * ws[k];
    ao += x * ws[DN + k];
  }
  for (int off = 16; off; off >>= 1) {
    as += __shfl_xor(as, off, 32);
    ao += __shfl_xor(ao, off, 32);
  }
  if (lane == 0) {
    s_self[row] = as + ws[256];
    s_other[row] = ao + ws[257];
  }
}

// ---------------- kernel 3: per-column softmax stats (axis=0) ---------------
__global__ __launch_bounds__(256) void stats_kernel(
    const float* __restrict__ s_self, const float* __restrict__ s_other,
    const int* __restrict__ group,
    float* __restrict__ Mp, float* __restrict__ iDp,
    float* __restrict__ Mn, float* __restrict__ iDn) {
  __shared__ float ss[NNODES];
  __shared__ int gg[NNODES];
  for (int t = threadIdx.x; t < NNODES; t += 256) {
    ss[t] = s_self[t];
    gg[t] = group[t];
  }
  __syncthreads();

  int col = blockIdx.x * 8 + (threadIdx.x >> 5);
  int lane = threadIdx.x & 31;
  float so = s_other[col];
  int gj = group[col];

  float mp = -INFINITY, mn = -INFINITY;
  for (int i = lane; i < NNODES; i += 32) {
    float e = ss[i] + so;
    bool same = (gg[i] == gj);
    float lp = leaky(same ? e : e * NEGC);
    float ln = leaky(same ? e * NEGC : e);
    mp = fmaxf(mp, lp);
    mn = fmaxf(mn, ln);
  }
  for (int off = 16; off; off >>= 1) {
    mp = fmaxf(mp, __shfl_xor(mp, off, 32));
    mn = fmaxf(mn, __shfl_xor(mn, off, 32));
  }
  float dp = 0.f, dq = 0.f;
  for (int i = lane; i < NNODES; i += 32) {
    float e = ss[i] + so;
    bool same = (gg[i] == gj);
    float lp = leaky(same ? e : e * NEGC);
    float ln = leaky(same ? e * NEGC : e);
    dp += __expf(lp - mp);
    dq += __expf(ln - mn);
  }
  for (int off = 16; off; off >>= 1) {
    dp += __shfl_xor(dp, off, 32);
    dq += __shfl_xor(dq, off, 32);
  }
  if (lane == 0) {
    Mp[col] = mp;
    iDp[col] = 1.f / dp;
    Mn[col] = mn;
    iDn[col] = 1.f / dq;
  }
}

// -------- kernel 4: out = (pos_alpha + neg_alpha) @ relu(node), f32 WMMA ----
// B panel staged via gfx1250 async-to-LDS (ASYNCcnt), double buffered.

#define CH 64                     // node rows per chunk
#define LDB 136                   // chunk row stride in floats (8*row+col banks)
#define NCHUNK (NNODES / CH)      // 16
#define CHUNKF (CH * LDB)         // floats per chunk buffer

__device__ __forceinline__ void async_wait8() {
#if __has_builtin(__builtin_amdgcn_s_wait_asynccnt)
  __builtin_amdgcn_s_wait_asynccnt(8);
#else
  asm volatile("s_wait_asynccnt 8" ::: "memory");
#endif
}
__device__ __forceinline__ void async_wait0() {
#if __has_builtin(__builtin_amdgcn_s_wait_asynccnt)
  __builtin_amdgcn_s_wait_asynccnt(0);
#else
  asm volatile("s_wait_asynccnt 0" ::: "memory");
#endif
}

// Issue one 32KB chunk (rows [64c,64c+64) x 128 cols) as async b128 copies.
// 256 threads x 8 instructions x 16B/lane -> 8 ASYNCcnt ticks per wave.
__device__ __forceinline__ void issue_chunk(const float* __restrict__ node,
                                            int c, const float* bufLds) {
  unsigned t = threadIdx.x;
  unsigned ldsBase = (unsigned)(unsigned long long)(size_t)bufLds;  // addr[31:0]
#pragma unroll
  for (int q = 0; q < 8; ++q) {
    unsigned u = t + 256u * q;   // 16-byte unit id, 0..2047
    unsigned row = u >> 5;       // 0..63
    unsigned uir = u & 31u;      // 16B unit within row
    const float* g = node + (size_t)(c * CH + row) * DN + uir * 4u;
    unsigned l = ldsBase + row * (LDB * 4u) + uir * 16u;
    asm volatile("global_load_async_to_lds_b128 %0, %1, off"
                 :: "v"(l), "v"(g)
                 : "memory");
  }
}

__global__ __launch_bounds__(256) void gemm_kernel(
    const float* __restrict__ node, const int* __restrict__ group,
    const float* __restrict__ s_self, const float* __restrict__ s_other,
    const float* __restrict__ Mp, const float* __restrict__ iDp,
    const float* __restrict__ Mn, const float* __restrict__ iDn,
    float* __restrict__ out) {
  // Row stride 1028 (== 4 mod 64 banks): the two half-wave K-pairs of the
  // WMMA A-frag land on disjoint bank sets -> conflict-free ds reads.
  constexpr int LDA = 1028;
  __shared__ float At[16 * LDA];
  __shared__ float Bs[2 * CHUNKF];

  int i0 = blockIdx.x * 16;

  // Kick off the first B chunk so the async copy overlaps the exp phase.
  issue_chunk(node, 0, Bs);

  // Phase 1: materialize the 16x1024 alpha-sum tile in LDS (2 exps / elem).
  for (int idx = threadIdx.x; idx < 16 * NNODES; idx += 256) {
    int il = idx >> 10;
    int j = idx & (NNODES - 1);
    float e = s_self[i0 + il] + s_other[j];
    bool same = (group[i0 + il] == group[j]);
    float lp = leaky(same ? e : e * NEGC);
    float ln = leaky(same ? e * NEGC : e);
    At[il * LDA + j] = __expf(lp - Mp[j]) * iDp[j] + __expf(ln - Mn[j]) * iDn[j];
  }
  __syncthreads();

  // Phase 2: wave w owns output columns [16w,16w+16); K-loop over j chunks.
  int lane = threadIdx.x & 31;
  int wave = threadIdx.x >> 5;
  int half = lane >> 4;   // 0: K pair {k,k+1}; 1: K pair {k+2,k+3}
  int nn = lane & 15;     // M for A-frag, N for B-frag
  int col = wave * 16 + nn;

  v8f acc0 = {};
  v8f acc1 = {};

  for (int c = 0; c < NCHUNK; ++c) {
    const float* B = Bs + (c & 1) * CHUNKF;
    if (c + 1 < NCHUNK) {
      issue_chunk(node, c + 1, Bs + ((c + 1) & 1) * CHUNKF);
      async_wait8();   // chunk c complete; chunk c+1 may stay in flight
    } else {
      async_wait0();
    }
    __syncthreads();   // chunk c visible to all waves

    int jbase = c * CH;
    for (int kl = 0; kl < CH; kl += 8) {
      int k0 = kl + 2 * half;
      v2f a0, b0;
      a0.x = At[nn * LDA + jbase + k0];
      a0.y = At[nn * LDA + jbase + k0 + 1];
      b0.x = fmaxf(B[k0 * LDB + col], 0.f);
      b0.y = fmaxf(B[(k0 + 1) * LDB + col], 0.f);
      acc0 = __builtin_amdgcn_wmma_f32_16x16x4_f32(false, a0, false, b0,
                                                   (short)0, acc0, false, false);
      int k1 = kl + 4 + 2 * half;
      v2f a1, b1;
      a1.x = At[nn * LDA + jbase + k1];
      a1.y = At[nn * LDA + jbase + k1 + 1];
      b1.x = fmaxf(B[k1 * LDB + col], 0.f);
      b1.y = fmaxf(B[(k1 + 1) * LDB + col], 0.f);
      acc1 = __builtin_amdgcn_wmma_f32_16x16x4_f32(false, a1, false, b1,
                                                   (short)0, acc1, false, false);
    }
    __syncthreads();   // chunk c buffer reusable for chunk c+2
  }

  // D layout (16x16 f32): VGPR r -> lanes 0-15: M=r, lanes 16-31: M=r+8.
  int orow = i0 + 8 * half;
#pragma unroll
  for (int r = 0; r < 8; ++r) {
    out[(orow + r) * DN + col] = acc0[r] + acc1[r];
  }
}

// ---------------------------------------------------------------------------
extern "C" void kernel_launch(void* const* d_in, const int* in_sizes, int n_in,
                              void* d_out, int out_size, void* d_ws, size_t ws_size,
                              hipStream_t stream) {
  const float* node = (const float*)d_in[0];
  const int* group = (const int*)d_in[1];
  const float* W_emb = (const float*)d_in[2];
  const float* b_emb = (const float*)d_in[3];
  const float* w_att = (const float*)d_in[4];
  const float* b_att = (const float*)d_in[5];
  float* ws = (float*)d_ws;
  float* out = (float*)d_out;

  float* s_self = ws + 512;
  float* s_other = ws + 1536;
  float* Mp = ws + 2560;
  float* iDp = ws + 3584;
  float* Mn = ws + 4608;
  float* iDn = ws + 5632;

  prep_kernel<<<1, 128, 0, stream>>>(W_emb, b_emb, w_att, b_att, ws);
  score_kernel<<<128, 256, 0, stream>>>(node, ws, s_self, s_other);
  stats_kernel<<<128, 256, 0, stream>>>(s_self, s_other, group, Mp, iDp, Mn, iDn);
  gemm_kernel<<<64, 256, 0, stream>>>(node, group, s_self, s_other,
                                      Mp, iDp, Mn, iDn, out);
}